// HarmonicFNOBlocks_30605936951522
// MI455X (gfx1250) — compile-verified
//
#include <hip/hip_runtime.h>
#include <hip/hip_bf16.h>
#include <math.h>

typedef float v2f __attribute__((ext_vector_type(2)));
typedef float v8f __attribute__((ext_vector_type(8)));

__device__ __forceinline__ v8f wmma4(v2f a, v2f b, v8f c) {
  // D = A(16x4) * B(4x16) + C, full fp32
  return __builtin_amdgcn_wmma_f32_16x16x4_f32(false, a, false, b, (short)0, c, false, false);
}
__device__ __forceinline__ v2f mk2(float a, float b) { v2f v; v.x = a; v.y = b; return v; }
__device__ __forceinline__ float gelu_t(float v) {
  float v3 = v * v * v;
  return 0.5f * v * (1.0f + tanhf(0.7978845608028654f * (v + 0.044715f * v3)));
}

// ---------------- problem sizes ----------------
// B=2, C=CO=32, N=64, MODES=16, RBINS=16, NL=9, HID=16, K144=144, M3=4096
constexpr int NVOL = 64;            // B*C = B*CO
constexpr size_t SZ_BASIS = (size_t)144 * 4096;
constexpr size_t SZ_OUT1  = (size_t)NVOL * 65536;  // [v][x][y][kz16]
constexpr size_t SZ_OUT2  = (size_t)NVOL * 16384;  // [v][x][ky16][kz16]
constexpr size_t SZ_XS    = (size_t)NVOL * 4096;   // [v][m]
constexpr size_t SZ_CF    = (size_t)144 * 64;      // [k][v]

constexpr size_t OFF_BASIS   = 0;
constexpr size_t OFF_DF_RE   = OFF_BASIS + SZ_BASIS;   // Df[kf][n] 16x64
constexpr size_t OFF_DF_IM   = OFF_DF_RE + 1024;
constexpr size_t OFF_DFT_RE  = OFF_DF_IM + 1024;       // DfT[n][kf] 64x16
constexpr size_t OFF_DFT_IM  = OFF_DFT_RE + 1024;
constexpr size_t OFF_DI_RE   = OFF_DFT_IM + 1024;      // Di[n][kf] 64x16 (incl 1/64)
constexpr size_t OFF_DI_IM   = OFF_DI_RE + 1024;
constexpr size_t OFF_OUT1_RE = OFF_DI_IM + 1024;
constexpr size_t OFF_OUT1_IM = OFF_OUT1_RE + SZ_OUT1;
constexpr size_t OFF_OUT2_RE = OFF_OUT1_IM + SZ_OUT1;
constexpr size_t OFF_OUT2_IM = OFF_OUT2_RE + SZ_OUT2;
constexpr size_t OFF_XS_RE   = OFF_OUT2_IM + SZ_OUT2;
constexpr size_t OFF_XS_IM   = OFF_XS_RE + SZ_XS;
constexpr size_t OFF_CF_RE   = OFF_XS_IM + SZ_XS;
constexpr size_t OFF_CF_IM   = OFF_CF_RE + SZ_CF;
constexpr size_t OFF_CF2_RE  = OFF_CF_IM + SZ_CF;
constexpr size_t OFF_CF2_IM  = OFF_CF2_RE + SZ_CF;
constexpr size_t OFF_XS2_RE  = OFF_CF2_IM + SZ_CF;
constexpr size_t OFF_XS2_IM  = OFF_XS2_RE + SZ_XS;
constexpr size_t OFF_INV1_RE = OFF_XS2_IM + SZ_XS;
constexpr size_t OFF_INV1_IM = OFF_INV1_RE + SZ_OUT2;
constexpr size_t OFF_INV2_RE = OFF_INV1_IM + SZ_OUT2;
constexpr size_t OFF_INV2_IM = OFF_INV2_RE + SZ_OUT1;

// ---------------- K0: build BASIS + DFT tables on device ----------------
__global__ void k_init(float* __restrict__ ws) {
  __shared__ int   cnt[16];
  __shared__ float wsh[16];
  const int t = threadIdx.x;  // 256 threads, single block
  if (t < 16) cnt[t] = 0;
  __syncthreads();
  const float rmax = sqrtf(192.0f);  // max |k| = 8 per axis
  for (int m = t; m < 4096; m += 256) {
    int ix = m >> 8, iy = (m >> 4) & 15, iz = m & 15;
    float kx = (ix < 8) ? (float)ix : (float)(ix - 16);
    float ky = (iy < 8) ? (float)iy : (float)(iy - 16);
    float kz = (iz < 8) ? (float)iz : (float)(iz - 16);
    float rr = sqrtf(kx * kx + ky * ky + kz * kz);
    int s = min((int)(rr / (rmax + 1e-6f) * 16.0f), 15);
    atomicAdd(&cnt[s], 1);
  }
  __syncthreads();
  if (t < 16) wsh[t] = rsqrtf(fmaxf((float)cnt[t], 1.0f));
  __syncthreads();
  for (size_t i = t; i < SZ_BASIS; i += 256) ws[OFF_BASIS + i] = 0.0f;
  __syncthreads();
  for (int m = t; m < 4096; m += 256) {
    int ix = m >> 8, iy = (m >> 4) & 15, iz = m & 15;
    float kx = (ix < 8) ? (float)ix : (float)(ix - 16);
    float ky = (iy < 8) ? (float)iy : (float)(iy - 16);
    float kz = (iz < 8) ? (float)iz : (float)(iz - 16);
    float rr = sqrtf(kx * kx + ky * ky + kz * kz);
    float nzm = (rr > 0.f) ? 1.f : 0.f;
    float inv = (rr > 0.f) ? 1.f / fmaxf(rr, 1e-9f) : 0.f;
    float ux = kx * inv, uy = ky * inv, uz = kz * inv;
    float Y[9];
    Y[0] = 0.282095f;
    Y[1] = 0.488603f * uy; Y[2] = 0.488603f * uz; Y[3] = 0.488603f * ux;
    Y[4] = 1.092548f * ux * uy; Y[5] = 1.092548f * uy * uz;
    Y[6] = 0.315392f * (3.f * uz * uz - nzm);
    Y[7] = 1.092548f * ux * uz; Y[8] = 0.546274f * (ux * ux - uy * uy);
    int s = min((int)(rr / (rmax + 1e-6f) * 16.0f), 15);
    float wgt = wsh[s];
#pragma unroll
    for (int l = 0; l < 9; ++l)
      ws[OFF_BASIS + (size_t)(s * 9 + l) * 4096 + m] = wgt * Y[l];
  }
  for (int i = t; i < 1024; i += 256) {
    int kf = i >> 6, n = i & 63;
    float f = (kf < 8) ? (float)kf : (float)(kf - 16);
    float ang = -6.283185307179586f * f * (float)n * (1.0f / 64.0f);
    float cr = cosf(ang), ci = sinf(ang);
    ws[OFF_DF_RE + kf * 64 + n] = cr;
    ws[OFF_DF_IM + kf * 64 + n] = ci;
    ws[OFF_DFT_RE + n * 16 + kf] = cr;
    ws[OFF_DFT_IM + n * 16 + kf] = ci;
    ws[OFF_DI_RE + n * 16 + kf] = cr * (1.0f / 64.0f);   // conj, 1/N per stage
    ws[OFF_DI_IM + n * 16 + kf] = -ci * (1.0f / 64.0f);
  }
}

// ---------------- K1: forward truncated DFT over z (real input) ----------------
// out1[v,x,y,kz] = sum_z x[v,x,y,z] * Df[kz,z]
// The 64 MB input stream is staged through LDS with fully-coalesced float4
// loads; WMMA A-fragments then come from LDS (padded rows -> no bank storm).
__global__ void k_fwd_z(const float* __restrict__ X, float* __restrict__ ws) {
  __shared__ float at[64 * 65];  // 64 (x,y) rows x 64 z, +1 pad per row
  const int v = blockIdx.x;
  const int t = threadIdx.x;
  const int w = t >> 5, lane = t & 31, r = lane & 15, h = lane >> 4;
  const int rt = blockIdx.y * 4 + w;  // global row-tile of 16 rows
  const float* src = X + (size_t)v * 262144 + (size_t)blockIdx.y * 4096;
  __builtin_prefetch(src + 4096, 0, 1);  // pull next tile toward L2
  for (int i = t; i < 1024; i += 128) {  // 1024 float4 = 64x64 floats
    int row = i >> 4, c4 = i & 15;
    const float4 val = ((const float4*)src)[i];
    float* dst = &at[row * 65 + c4 * 4];
    dst[0] = val.x; dst[1] = val.y; dst[2] = val.z; dst[3] = val.w;
  }
  __syncthreads();
  const float* A  = &at[(w * 16 + r) * 65];
  const float* Br = ws + OFF_DFT_RE;
  const float* Bi = ws + OFF_DFT_IM;
  v8f accR = {}, accI = {};
#pragma unroll
  for (int k0 = 0; k0 < 64; k0 += 4) {
    int k = k0 + 2 * h;
    v2f a  = mk2(A[k], A[k + 1]);
    v2f br = mk2(Br[k * 16 + r], Br[(k + 1) * 16 + r]);
    v2f bi = mk2(Bi[k * 16 + r], Bi[(k + 1) * 16 + r]);
    accR = wmma4(a, br, accR);
    accI = wmma4(a, bi, accI);
  }
  float* Dr = ws + OFF_OUT1_RE + (size_t)v * 65536 + (size_t)rt * 256;
  float* Di = ws + OFF_OUT1_IM + (size_t)v * 65536 + (size_t)rt * 256;
#pragma unroll
  for (int p = 0; p < 8; ++p) {
    Dr[(p + 8 * h) * 16 + r] = accR[p];
    Di[(p + 8 * h) * 16 + r] = accI[p];
  }
}

// ---------------- K2: forward DFT over y (complex) ----------------
// out2[v,x,ky,kz] = sum_y Df[ky,y] * out1[v,x,y,kz]
__global__ void k_fwd_y(float* __restrict__ ws) {
  const int v = blockIdx.x;
  const int w = threadIdx.x >> 5, lane = threadIdx.x & 31, r = lane & 15, h = lane >> 4;
  const int xx = blockIdx.y * 4 + w;  // 0..63
  const float* Ar = ws + OFF_DF_RE;
  const float* Ai = ws + OFF_DF_IM;
  const float* Br = ws + OFF_OUT1_RE + (size_t)v * 65536 + (size_t)xx * 1024;
  const float* Bi = ws + OFF_OUT1_IM + (size_t)v * 65536 + (size_t)xx * 1024;
  v8f RR = {}, II = {}, RI = {}, IR = {};
#pragma unroll
  for (int k0 = 0; k0 < 64; k0 += 4) {
    int k = k0 + 2 * h;
    v2f ar = mk2(Ar[r * 64 + k], Ar[r * 64 + k + 1]);
    v2f ai = mk2(Ai[r * 64 + k], Ai[r * 64 + k + 1]);
    v2f br = mk2(Br[k * 16 + r], Br[(k + 1) * 16 + r]);
    v2f bi = mk2(Bi[k * 16 + r], Bi[(k + 1) * 16 + r]);
    RR = wmma4(ar, br, RR);
    II = wmma4(ai, bi, II);
    RI = wmma4(ar, bi, RI);
    IR = wmma4(ai, br, IR);
  }
  float* Dr = ws + OFF_OUT2_RE + (size_t)v * 16384 + (size_t)xx * 256;
  float* Di = ws + OFF_OUT2_IM + (size_t)v * 16384 + (size_t)xx * 256;
#pragma unroll
  for (int p = 0; p < 8; ++p) {
    Dr[(p + 8 * h) * 16 + r] = RR[p] - II[p];
    Di[(p + 8 * h) * 16 + r] = RI[p] + IR[p];
  }
}

// ---------------- K3: forward DFT over x (complex) ----------------
// xs[v, kx*256 + n] = sum_x Df[kx,x] * out2[v,x,n]   (n = ky*16+kz)
__global__ void k_fwd_x(float* __restrict__ ws) {
  const int v = blockIdx.x;
  const int w = threadIdx.x >> 5, lane = threadIdx.x & 31, r = lane & 15, h = lane >> 4;
  const int nt = blockIdx.y * 4 + w;  // 0..15
  const float* Ar = ws + OFF_DF_RE;
  const float* Ai = ws + OFF_DF_IM;
  const float* Br = ws + OFF_OUT2_RE + (size_t)v * 16384 + nt * 16;
  const float* Bi = ws + OFF_OUT2_IM + (size_t)v * 16384 + nt * 16;
  v8f RR = {}, II = {}, RI = {}, IR = {};
#pragma unroll
  for (int k0 = 0; k0 < 64; k0 += 4) {
    int k = k0 + 2 * h;
    v2f ar = mk2(Ar[r * 64 + k], Ar[r * 64 + k + 1]);
    v2f ai = mk2(Ai[r * 64 + k], Ai[r * 64 + k + 1]);
    v2f br = mk2(Br[k * 256 + r], Br[(k + 1) * 256 + r]);
    v2f bi = mk2(Bi[k * 256 + r], Bi[(k + 1) * 256 + r]);
    RR = wmma4(ar, br, RR);
    II = wmma4(ai, bi, II);
    RI = wmma4(ar, bi, RI);
    IR = wmma4(ai, br, IR);
  }
  float* Dr = ws + OFF_XS_RE + (size_t)v * 4096 + nt * 16;
  float* Di = ws + OFF_XS_IM + (size_t)v * 4096 + nt * 16;
#pragma unroll
  for (int p = 0; p < 8; ++p) {
    Dr[(p + 8 * h) * 256 + r] = RR[p] - II[p];
    Di[(p + 8 * h) * 256 + r] = RI[p] + IR[p];
  }
}

// ---------------- K4: cf[k][v] = sum_m BASIS[k][m] * xs[v][m]  (K=4096) ----------------
__global__ void k_basis_fwd(float* __restrict__ ws) {
  const int kt = blockIdx.x;  // 0..8 (144 rows)
  const int vt = blockIdx.y;  // 0..3 (64 vols)
  const int lane = threadIdx.x & 31, r = lane & 15, h = lane >> 4;
  const float* A  = ws + OFF_BASIS + (size_t)(kt * 16 + r) * 4096;
  const float* Br = ws + OFF_XS_RE + (size_t)(vt * 16 + r) * 4096;
  const float* Bi = ws + OFF_XS_IM + (size_t)(vt * 16 + r) * 4096;
  v8f aR = {}, aI = {};
#pragma unroll 4
  for (int k0 = 0; k0 < 4096; k0 += 4) {
    int k = k0 + 2 * h;
    v2f a  = mk2(A[k], A[k + 1]);
    v2f br = mk2(Br[k], Br[k + 1]);
    v2f bi = mk2(Bi[k], Bi[k + 1]);
    aR = wmma4(a, br, aR);
    aI = wmma4(a, bi, aI);
  }
  float* Dr = ws + OFF_CF_RE + (size_t)kt * 16 * 64 + vt * 16;
  float* Di = ws + OFF_CF_IM + (size_t)kt * 16 * 64 + vt * 16;
#pragma unroll
  for (int p = 0; p < 8; ++p) {
    Dr[(p + 8 * h) * 64 + r] = aR[p];
    Di[(p + 8 * h) * 64 + r] = aI[p];
  }
}

// ---------------- K5: per-(s,l) complex 32x32 channel mix (tiny, VALU) ----------------
__global__ void k_spec(const float* __restrict__ wr, const float* __restrict__ wi,
                       float* __restrict__ ws) {
  const int k = blockIdx.x;   // 0..143
  const int t = threadIdx.x;  // 0..63 : b = t>>5, o = t&31
  const int bb = t >> 5, o = t & 31;
  const float* cr = ws + OFF_CF_RE + (size_t)k * 64 + bb * 32;
  const float* ci = ws + OFF_CF_IM + (size_t)k * 64 + bb * 32;
  float sr = 0.f, si = 0.f;
#pragma unroll 4
  for (int i = 0; i < 32; ++i) {
    float ar = cr[i], ai = ci[i];
    float br = wr[((size_t)k * 32 + i) * 32 + o];
    float bi = wi[((size_t)k * 32 + i) * 32 + o];
    sr += ar * br - ai * bi;
    si += ar * bi + ai * br;
  }
  ws[OFF_CF2_RE + (size_t)k * 64 + t] = sr;
  ws[OFF_CF2_IM + (size_t)k * 64 + t] = si;
}

// ---------------- K6: xs2[v][m] = sum_k cf2[k][v] * BASIS[k][m]  (K=144) ----------------
__global__ void k_basis_inv(float* __restrict__ ws) {
  const int vt = blockIdx.x;  // 0..3
  const int w = threadIdx.x >> 5, lane = threadIdx.x & 31, r = lane & 15, h = lane >> 4;
  const int mt = blockIdx.y * 4 + w;  // 0..255
  const float* Ar = ws + OFF_CF2_RE;
  const float* Ai = ws + OFF_CF2_IM;
  const float* B  = ws + OFF_BASIS + mt * 16;
  v8f aR = {}, aI = {};
#pragma unroll
  for (int k0 = 0; k0 < 144; k0 += 4) {
    int k = k0 + 2 * h;
    v2f ar = mk2(Ar[k * 64 + vt * 16 + r], Ar[(k + 1) * 64 + vt * 16 + r]);
    v2f ai = mk2(Ai[k * 64 + vt * 16 + r], Ai[(k + 1) * 64 + vt * 16 + r]);
    v2f b  = mk2(B[(size_t)k * 4096 + r], B[(size_t)(k + 1) * 4096 + r]);
    aR = wmma4(ar, b, aR);
    aI = wmma4(ai, b, aI);
  }
  float* Dr = ws + OFF_XS2_RE + (size_t)vt * 16 * 4096 + mt * 16;
  float* Di = ws + OFF_XS2_IM + (size_t)vt * 16 * 4096 + mt * 16;
#pragma unroll
  for (int p = 0; p < 8; ++p) {
    Dr[(size_t)(p + 8 * h) * 4096 + r] = aR[p];
    Di[(size_t)(p + 8 * h) * 4096 + r] = aI[p];
  }
}

// ---------------- K7: inverse DFT over kx ----------------
// inv1[v,x,n] = sum_kx Di[x,kx] * xs2[v, kx*256+n]
__global__ void k_inv_x(float* __restrict__ ws) {
  const int v = blockIdx.x;
  const int w = threadIdx.x >> 5, lane = threadIdx.x & 31, r = lane & 15, h = lane >> 4;
  const int idx = blockIdx.y * 4 + w;  // 0..63
  const int xt = idx >> 4, nt = idx & 15;
  const float* Ar = ws + OFF_DI_RE;
  const float* Ai = ws + OFF_DI_IM;
  const float* Br = ws + OFF_XS2_RE + (size_t)v * 4096 + nt * 16;
  const float* Bi = ws + OFF_XS2_IM + (size_t)v * 4096 + nt * 16;
  v8f RR = {}, II = {}, RI = {}, IR = {};
#pragma unroll
  for (int k0 = 0; k0 < 16; k0 += 4) {
    int k = k0 + 2 * h;
    v2f ar = mk2(Ar[(xt * 16 + r) * 16 + k], Ar[(xt * 16 + r) * 16 + k + 1]);
    v2f ai = mk2(Ai[(xt * 16 + r) * 16 + k], Ai[(xt * 16 + r) * 16 + k + 1]);
    v2f br = mk2(Br[k * 256 + r], Br[(k + 1) * 256 + r]);
    v2f bi = mk2(Bi[k * 256 + r], Bi[(k + 1) * 256 + r]);
    RR = wmma4(ar, br, RR);
    II = wmma4(ai, bi, II);
    RI = wmma4(ar, bi, RI);
    IR = wmma4(ai, br, IR);
  }
  float* Dr = ws + OFF_INV1_RE + (size_t)v * 16384 + (size_t)xt * 4096 + nt * 16;
  float* Di = ws + OFF_INV1_IM + (size_t)v * 16384 + (size_t)xt * 4096 + nt * 16;
#pragma unroll
  for (int p = 0; p < 8; ++p) {
    Dr[(p + 8 * h) * 256 + r] = RR[p] - II[p];
    Di[(p + 8 * h) * 256 + r] = RI[p] + IR[p];
  }
}

// ---------------- K8: inverse DFT over ky ----------------
// inv2[v,x,y,kz] = sum_ky Di[y,ky] * inv1[v,x,ky,kz]
__global__ void k_inv_y(float* __restrict__ ws) {
  const int v = blockIdx.x, xx = blockIdx.y;
  const int w = threadIdx.x >> 5, lane = threadIdx.x & 31, r = lane & 15, h = lane >> 4;
  const int yt = w;  // 4 y-tiles
  const float* Ar = ws + OFF_DI_RE;
  const float* Ai = ws + OFF_DI_IM;
  const float* Br = ws + OFF_INV1_RE + (size_t)v * 16384 + (size_t)xx * 256;
  const float* Bi = ws + OFF_INV1_IM + (size_t)v * 16384 + (size_t)xx * 256;
  v8f RR = {}, II = {}, RI = {}, IR = {};
#pragma unroll
  for (int k0 = 0; k0 < 16; k0 += 4) {
    int k = k0 + 2 * h;
    v2f ar = mk2(Ar[(yt * 16 + r) * 16 + k], Ar[(yt * 16 + r) * 16 + k + 1]);
    v2f ai = mk2(Ai[(yt * 16 + r) * 16 + k], Ai[(yt * 16 + r) * 16 + k + 1]);
    v2f br = mk2(Br[k * 16 + r], Br[(k + 1) * 16 + r]);
    v2f bi = mk2(Bi[k * 16 + r], Bi[(k + 1) * 16 + r]);
    RR = wmma4(ar, br, RR);
    II = wmma4(ai, bi, II);
    RI = wmma4(ar, bi, RI);
    IR = wmma4(ai, br, IR);
  }
  float* Dr = ws + OFF_INV2_RE + (size_t)v * 65536 + (size_t)xx * 1024 + (size_t)yt * 256;
  float* Di = ws + OFF_INV2_IM + (size_t)v * 65536 + (size_t)xx * 1024 + (size_t)yt * 256;
#pragma unroll
  for (int p = 0; p < 8; ++p) {
    Dr[(p + 8 * h) * 16 + r] = RR[p] - II[p];
    Di[(p + 8 * h) * 16 + r] = RI[p] + IR[p];
  }
}

// ---------------- K9: fused inverse-DFT(z) + skips + complex MLP epilogue ----------------
__global__ void k_epilogue(const float* __restrict__ X, const float* __restrict__ ws,
                           const float* __restrict__ wfr, const float* __restrict__ wfi,
                           const float* __restrict__ bfr, const float* __restrict__ bfi,
                           const float* __restrict__ wgr,
                           const float* __restrict__ w1r, const float* __restrict__ w1i,
                           const float* __restrict__ b1r, const float* __restrict__ b1i,
                           const float* __restrict__ w2r, const float* __restrict__ w2i,
                           const float* __restrict__ b2r, float* __restrict__ out) {
  __shared__ float xrow[32 * 64];               // x[b,c,x,y,:]   [c][z]
  __shared__ float sre[32 * 16], sim[32 * 16];  // inv2 slice     [o][kz]
  __shared__ float x1r[64 * 32], x1i[64 * 32];  // x1             [z][o]
  __shared__ float hre[64 * 16], him[64 * 16];  // hidden         [z][hd]
  const int b = blockIdx.x, xx = blockIdx.y, yy = blockIdx.z;
  const int t = threadIdx.x;
  const int w = t >> 5, lane = t & 31, r = lane & 15, h = lane >> 4;
  const size_t xbase = (size_t)b * 32 * 262144 + (size_t)xx * 4096 + (size_t)yy * 64;
  for (int i = t; i < 2048; i += 256) {
    int c = i >> 6, z = i & 63;
    xrow[i] = X[xbase + (size_t)c * 262144 + z];
  }
  for (int i = t; i < 512; i += 256) {
    int o = i >> 4, kz = i & 15;
    size_t idx = (size_t)(b * 32 + o) * 65536 + (size_t)xx * 1024 + (size_t)yy * 16 + kz;
    sre[i] = ws[OFF_INV2_RE + idx];
    sim[i] = ws[OFF_INV2_IM + idx];
  }
  __syncthreads();
  {  // x1[z][o] = iDFT_z(inv2) + Wfno @ x + b_fno   (8 waves: 4 z-tiles x 2 o-tiles)
    const int zt = w >> 1, ot = w & 1;
    const float* DiR = ws + OFF_DI_RE;
    const float* DiI = ws + OFF_DI_IM;
    v8f aP = {}, aN = {}, aI = {};
#pragma unroll
    for (int k0 = 0; k0 < 16; k0 += 4) {  // K over kz
      int k = k0 + 2 * h;
      v2f dr = mk2(DiR[(zt * 16 + r) * 16 + k], DiR[(zt * 16 + r) * 16 + k + 1]);
      v2f di = mk2(DiI[(zt * 16 + r) * 16 + k], DiI[(zt * 16 + r) * 16 + k + 1]);
      v2f br = mk2(sre[(ot * 16 + r) * 16 + k], sre[(ot * 16 + r) * 16 + k + 1]);
      v2f bi = mk2(sim[(ot * 16 + r) * 16 + k], sim[(ot * 16 + r) * 16 + k + 1]);
      aP = wmma4(dr, br, aP);
      aN = wmma4(di, bi, aN);
      aI = wmma4(dr, bi, aI);
      aI = wmma4(di, br, aI);
    }
#pragma unroll
    for (int k0 = 0; k0 < 32; k0 += 4) {  // K over input channels c
      int k = k0 + 2 * h;
      v2f a  = mk2(xrow[k * 64 + zt * 16 + r], xrow[(k + 1) * 64 + zt * 16 + r]);
      v2f br = mk2(wfr[(ot * 16 + r) * 32 + k], wfr[(ot * 16 + r) * 32 + k + 1]);
      v2f bi = mk2(wfi[(ot * 16 + r) * 32 + k], wfi[(ot * 16 + r) * 32 + k + 1]);
      aP = wmma4(a, br, aP);
      aI = wmma4(a, bi, aI);
    }
    int o = ot * 16 + r;
#pragma unroll
    for (int p = 0; p < 8; ++p) {
      int z = zt * 16 + p + 8 * h;
      x1r[z * 32 + o] = aP[p] - aN[p] + bfr[o];
      x1i[z * 32 + o] = aI[p] + bfi[o];
    }
  }
  __syncthreads();
  if (w < 4) {  // h = cgelu(W1 @ x1 + b1)   (4 waves: 4 z-tiles, HID=16 => 1 n-tile)
    v8f hP = {}, hN = {}, hI = {};
#pragma unroll
    for (int k0 = 0; k0 < 32; k0 += 4) {  // K over o channels
      int k = k0 + 2 * h;
      v2f ar = mk2(x1r[(w * 16 + r) * 32 + k], x1r[(w * 16 + r) * 32 + k + 1]);
      v2f ai = mk2(x1i[(w * 16 + r) * 32 + k], x1i[(w * 16 + r) * 32 + k + 1]);
      v2f br = mk2(w1r[r * 32 + k], w1r[r * 32 + k + 1]);
      v2f bi = mk2(w1i[r * 32 + k], w1i[r * 32 + k + 1]);
      hP = wmma4(ar, br, hP);
      hN = wmma4(ai, bi, hN);
      hI = wmma4(ar, bi, hI);
      hI = wmma4(ai, br, hI);
    }
#pragma unroll
    for (int p = 0; p < 8; ++p) {
      int z = w * 16 + p + 8 * h;
      hre[z * 16 + r] = gelu_t(hP[p] - hN[p] + b1r[r]);
      him[z * 16 + r] = gelu_t(hI[p] + b1i[r]);
    }
  }
  __syncthreads();
  {  // out = Re(W2 @ h) + b2_r + x * wgate_r   (8 waves)
    const int zt = w >> 1, ot = w & 1;
    v8f yP = {}, yN = {};
#pragma unroll
    for (int k0 = 0; k0 < 16; k0 += 4) {  // K over HID
      int k = k0 + 2 * h;
      v2f ar = mk2(hre[(zt * 16 + r) * 16 + k], hre[(zt * 16 + r) * 16 + k + 1]);
      v2f ai = mk2(him[(zt * 16 + r) * 16 + k], him[(zt * 16 + r) * 16 + k + 1]);
      v2f br = mk2(w2r[(ot * 16 + r) * 16 + k], w2r[(ot * 16 + r) * 16 + k + 1]);
      v2f bi = mk2(w2i[(ot * 16 + r) * 16 + k], w2i[(ot * 16 + r) * 16 + k + 1]);
      yP = wmma4(ar, br, yP);
      yN = wmma4(ai, bi, yN);
    }
    int o = ot * 16 + r;
    float g = wgr[o];
#pragma unroll
    for (int p = 0; p < 8; ++p) {
      int z = zt * 16 + p + 8 * h;
      out[(size_t)(b * 32 + o) * 262144 + (size_t)xx * 4096 + (size_t)yy * 64 + z] =
          yP[p] - yN[p] + b2r[o] + xrow[o * 64 + z] * g;
    }
  }
}

extern "C" void kernel_launch(void* const* d_in, const int* in_sizes, int n_in,
                              void* d_out, int out_size, void* d_ws, size_t ws_size,
                              hipStream_t stream) {
  (void)in_sizes; (void)n_in; (void)out_size; (void)ws_size;
  const float* X   = (const float*)d_in[0];
  const float* wfr = (const float*)d_in[1];
  const float* wfi = (const float*)d_in[2];
  const float* bfr = (const float*)d_in[3];
  const float* bfi = (const float*)d_in[4];
  const float* wgr = (const float*)d_in[5];
  /* d_in[6] w_gate_i: does not affect real part of output */
  const float* wsr = (const float*)d_in[7];
  const float* wsi = (const float*)d_in[8];
  const float* w1r = (const float*)d_in[9];
  const float* w1i = (const float*)d_in[10];
  const float* b1r = (const float*)d_in[11];
  const float* b1i = (const float*)d_in[12];
  const float* w2r = (const float*)d_in[13];
  const float* w2i = (const float*)d_in[14];
  const float* b2r = (const float*)d_in[15];
  /* d_in[16] b2_i: imaginary bias dropped by .real */
  float* out = (float*)d_out;
  float* ws  = (float*)d_ws;

  k_init     <<<1, 256, 0, stream>>>(ws);
  k_fwd_z    <<<dim3(64, 64), 128, 0, stream>>>(X, ws);
  k_fwd_y    <<<dim3(64, 16), 128, 0, stream>>>(ws);
  k_fwd_x    <<<dim3(64, 4), 128, 0, stream>>>(ws);
  k_basis_fwd<<<dim3(9, 4), 32, 0, stream>>>(ws);
  k_spec     <<<144, 64, 0, stream>>>(wsr, wsi, ws);
  k_basis_inv<<<dim3(4, 64), 128, 0, stream>>>(ws);
  k_inv_x    <<<dim3(64, 16), 128, 0, stream>>>(ws);
  k_inv_y    <<<dim3(64, 64), 128, 0, stream>>>(ws);
  k_epilogue <<<dim3(2, 64, 64), 256, 0, stream>>>(X, ws, wfr, wfi, bfr, bfi, wgr,
                                                   w1r, w1i, b1r, b1i, w2r, w2i, b2r, out);
}